// LP_21844203668398
// MI455X (gfx1250) — compile-verified
//
#include <hip/hip_runtime.h>
#include <hip/hip_bf16.h>

typedef __attribute__((ext_vector_type(2))) float v2f;
typedef __attribute__((ext_vector_type(8))) float v8f;

#define ALPHA_F 0.5f
#define NCLS 16

// ---------------------------------------------------------------------------
// Prep: hs = W[node] * x, agg = 0   (only before layer 0)
// ---------------------------------------------------------------------------
__global__ __launch_bounds__(256)
void lp_prep(const float* __restrict__ x, const float* __restrict__ W,
             float* __restrict__ hs, float* __restrict__ agg, int total) {
  int i = blockIdx.x * 256 + threadIdx.x;
  if (i < total) {
    hs[i]  = W[i >> 4] * x[i];
    agg[i] = 0.0f;
  }
}

// ---------------------------------------------------------------------------
// Edge phase: 16 lanes per edge (one class per lane).
// Gather hs[src] (64B coalesced), scale by edge weight, scatter-add into agg
// via 16 contiguous global f32 atomics. Everything stays L2-resident.
// ---------------------------------------------------------------------------
__global__ __launch_bounds__(256)
void lp_edge(const int* __restrict__ ei, const float* __restrict__ ew,
             const float* __restrict__ hs, float* agg, int n_edges) {
  int gid = blockIdx.x * 256 + threadIdx.x;
  int e = gid >> 4;
  int c = gid & (NCLS - 1);
  if (e < n_edges) {
    if (c == 0) {
      // stream-ahead hints: emits global_prefetch_b8
      __builtin_prefetch(ei + e + 4096, 0, 1);
      __builtin_prefetch(ew + e + 4096, 0, 1);
    }
    int   src = ei[e];
    int   dst = ei[n_edges + e];
    float w   = ew[e];
    float v   = hs[src * NCLS + c] * w;
    atomicAdd(&agg[dst * NCLS + c], v);
  }
}

// ---------------------------------------------------------------------------
// Node phase: per wave, one 16x16 f32 tile (16 nodes x 16 classes).
//   U = diag((1-a)*deg^2) x AGG + a*H       -> 4 chained v_wmma_f32_16x16x4_f32
//   out = U / rowsum(U)                      -> shfl_xor reduce per half-wave
// Fuses: write h_out, write hs = W*h_out for next layer's gather, clear agg.
//
// Layouts (ISA 7.12.2, f32):
//   C/D: VGPR r, lanes 0-15 -> (M=r, N=lane); lanes 16-31 -> (M=r+8, N=lane-16)
//   A 16x4 chunk: V0: K=0 (lo half)/K=2 (hi half); V1: K=1 / K=3
//   B 4x16 chunk: V0: K=0 (lo)/K=2 (hi); V1: K=1 / K=3, N=lane%16
// Contraction K = node index within tile (A is diagonal), M = node, N = class.
// ---------------------------------------------------------------------------
__global__ __launch_bounds__(256)
void lp_node(const float* __restrict__ h_in, float* agg,
             const float* __restrict__ W, const float* __restrict__ degree,
             float* __restrict__ h_out, float* __restrict__ hs_out,
             int n_tiles) {
  const int lane  = threadIdx.x & 31;
  const int wave  = threadIdx.x >> 5;
  const int tile  = blockIdx.x * 8 + wave;
  if (tile >= n_tiles) return;               // wave-uniform: EXEC stays all-1s
  const int node0 = tile * 16;
  const int m     = lane & 15;               // lane's row (A) / column (B,C,D)
  const int hi    = (lane >= 16) ? 1 : 0;

  // diagonal value for this lane's row
  float dg = degree[node0 + m];
  float av = (1.0f - ALPHA_F) * dg * dg;

  // C = ALPHA * H in C/D layout
  v8f acc;
#pragma unroll
  for (int r = 0; r < 8; ++r) {
    int row = node0 + r + (hi ? 8 : 0);
    acc[r] = ALPHA_F * h_in[row * NCLS + m];
  }

  // chain of four 16x16x4 f32 WMMAs: acc += diag(av) x AGG
#pragma unroll
  for (int q = 0; q < 4; ++q) {
    int k0 = 4 * q + 2 * hi;                 // this half-wave's K pair
    v2f a, b;
    a.x = (m == k0)     ? av : 0.0f;         // A[m][k0]
    a.y = (m == k0 + 1) ? av : 0.0f;         // A[m][k0+1]
    b.x = agg[(node0 + k0)     * NCLS + m];  // B[k0][m]   = AGG row k0
    b.y = agg[(node0 + k0 + 1) * NCLS + m];  // B[k0+1][m]
    acc = __builtin_amdgcn_wmma_f32_16x16x4_f32(
        /*neg_a=*/false, a, /*neg_b=*/false, b,
        /*c_mod=*/(short)0, acc, /*reuse_a=*/false, /*reuse_b=*/false);
  }

  // Row-normalize + fused outputs. After WMMA, VGPR r of lanes 0-15 holds row
  // r across classes; summing across each 16-lane half puts rowsum(r) in the
  // exact lane/VGPR that needs it as divisor.
#pragma unroll
  for (int r = 0; r < 8; ++r) {
    float s = acc[r];
    s += __shfl_xor(s, 1, 32);
    s += __shfl_xor(s, 2, 32);
    s += __shfl_xor(s, 4, 32);
    s += __shfl_xor(s, 8, 32);
    int   row = node0 + r + (hi ? 8 : 0);
    float o   = acc[r] / s;
    h_out [row * NCLS + m] = o;
    hs_out[row * NCLS + m] = W[row] * o;     // pre-scale for next edge phase
    agg   [row * NCLS + m] = 0.0f;           // reset accumulator for next layer
  }
}

// ---------------------------------------------------------------------------
extern "C" void kernel_launch(void* const* d_in, const int* in_sizes, int n_in,
                              void* d_out, int out_size, void* d_ws, size_t ws_size,
                              hipStream_t stream) {
  const float* x   = (const float*)d_in[0];   // (N, 16)
  const float* W   = (const float*)d_in[1];   // (1, N) -> per-node scale
  const float* ew  = (const float*)d_in[2];   // (E,)
  const float* deg = (const float*)d_in[3];   // (N,)
  const int*   ei  = (const int*)  d_in[4];   // (2, E): [0..E)=src, [E..2E)=dst

  const int N       = in_sizes[3];
  const int N_EDGES = in_sizes[2];
  const int total   = N * NCLS;
  const int n_tiles = N / 16;                 // 100000 / 16 = 6250 exact

  float* out = (float*)d_out;
  float* agg = (float*)d_ws;                  // N*16 f32
  float* hA  = agg + (size_t)total;           // N*16 f32 (h ping buffer)
  float* hs  = hA  + (size_t)total;           // N*16 f32 (W-scaled features)

  const int prep_blocks = (total + 255) / 256;
  const int edge_blocks = (N_EDGES * NCLS + 255) / 256;
  const int node_blocks = (n_tiles + 7) / 8;

  lp_prep<<<prep_blocks, 256, 0, stream>>>(x, W, hs, agg, total);

  for (int k = 0; k < 4; ++k) {
    // h ping-pong: x -> hA -> out -> hA -> out (layer 4 lands in d_out)
    const float* hin  = (k == 0) ? x : ((k & 1) ? hA : out);
    float*       hout = (k & 1) ? out : hA;
    lp_edge<<<edge_blocks, 256, 0, stream>>>(ei, ew, hs, agg, N_EDGES);
    lp_node<<<node_blocks, 256, 0, stream>>>(hin, agg, W, deg, hout, hs, n_tiles);
  }
}